// MixIT_Loss_28595892256860
// MI455X (gfx1250) — compile-verified
//
#include <hip/hip_runtime.h>
#include <math.h>

typedef float v2f __attribute__((ext_vector_type(2)));
typedef float v8f __attribute__((ext_vector_type(8)));

#define TAU_   1e-6f
#define BATCH  8
#define MSRC   8
#define NSIG   10      // 2 mixtures + 8 sources
#define WAVES  8
#define TPB    256
#define SPLIT  8       // T-chunks (blocks) per batch sample -> 64 blocks total

// ---------------------------------------------------------------------------
// Stage 1: 64 blocks (BATCH * SPLIT), 8 waves each. Every wave accumulates a
// partial 16x16 f32 Gram of the stacked signals [x0, x1, s0..s7, pad] over a
// contiguous 500-sample run using V_WMMA_F32_16X16X4_F32. For a symmetric
// Gram S*S^T the A (16x4) and B (4x16) per-lane register images are
// identical, so one float2 load per lane feeds both operands.
// Each block reduces its 8 wave-partials in LDS and emits one 256-float
// partial Gram to d_ws.
// ---------------------------------------------------------------------------
__global__ __launch_bounds__(TPB) void mixit_gram_kernel(
    const float* __restrict__ est,   // (B, M, T)
    const float* __restrict__ mix,   // (B, 2, T)
    float* __restrict__ partials,    // (BATCH*SPLIT, 256)
    int T)
{
    __shared__ float wgram[WAVES][256];

    const int b     = blockIdx.x / SPLIT;
    const int chunk = blockIdx.x % SPLIT;
    const int tid   = threadIdx.x;
    const int lane  = tid & 31;
    const int wave  = tid >> 5;
    const int sig   = lane & 15;            // signal row this lane feeds
    const int khalf = (lane >> 4) << 1;     // lanes 16-31 carry K=2,3

    const float* mix_b = mix + (size_t)b * 2 * T;
    const float* est_b = est + (size_t)b * MSRC * T;

    // Row pointer per lane; lanes 10..15 alias mixture row 0 (their Gram
    // rows/cols are never consumed; duplicate addresses hit cache for free).
    const float* row;
    if (sig < 2)          row = mix_b + (size_t)sig * T;
    else if (sig < NSIG)  row = est_b + (size_t)(sig - 2) * T;
    else                  row = mix_b;

    const int t_per_wave = T / (SPLIT * WAVES);       // 500
    const int iters = t_per_wave >> 2;                // 125 WMMAs per wave
    const float* p = row + (chunk * WAVES + wave) * t_per_wave + khalf;

    v8f acc = {};
    #pragma unroll 4
    for (int i = 0; i < iters; ++i) {
        v2f a;
        a.x = p[0];                          // K = khalf
        a.y = p[1];                          // K = khalf + 1
        acc = __builtin_amdgcn_wmma_f32_16x16x4_f32(
            /*neg_a=*/false, a, /*neg_b=*/false, a,
            /*c_mod=*/(short)0, acc, /*reuse_a=*/false, /*reuse_b=*/false);
        p += 4;
    }

    // C/D layout: VGPR r -> row = r + (lane>=16 ? 8 : 0), col = lane & 15
    const int rbase = (lane >> 4) << 3;
    #pragma unroll
    for (int r = 0; r < 8; ++r)
        wgram[wave][(rbase + r) * 16 + sig] = acc[r];
    __syncthreads();

    // Deterministic cross-wave reduction; thread tid owns Gram element tid.
    float g = 0.f;
    #pragma unroll
    for (int w = 0; w < WAVES; ++w) g += wgram[w][tid];
    partials[(size_t)blockIdx.x * 256 + tid] = g;
}

// Evaluate one two-group partition mask against a 16x16 Gram in LDS.
__device__ __forceinline__ float combo_loss(const float* gram, int mask)
{
    const float E0 = gram[0 * 16 + 0];
    const float E1 = gram[1 * 16 + 1];
    float n0 = E0, n1 = E1;
    #pragma unroll
    for (int m = 0; m < MSRC; ++m) {
        const bool in1 = (mask >> m) & 1;
        const float c0 = gram[0 * 16 + 2 + m];
        const float c1 = gram[1 * 16 + 2 + m];
        if (in1) n1 -= 2.f * c1; else n0 -= 2.f * c0;
        #pragma unroll
        for (int mm = 0; mm < MSRC; ++mm) {
            const bool in1b = (mask >> mm) & 1;
            const float gmm = gram[(2 + m) * 16 + 2 + mm];
            if (in1 && in1b)   n1 += gmm;
            if (!in1 && !in1b) n0 += gmm;
        }
    }
    const float s0 = -10.f * log10f(E0 / (n0 + TAU_ * E0));
    const float s1 = -10.f * log10f(E1 / (n1 + TAU_ * E1));
    return s0 + s1;
}

// ---------------------------------------------------------------------------
// Stage 2: one block. Per batch sample: reduce the SPLIT partial Grams,
// evaluate all 254 partitions (one mask per thread), block-min, accumulate
// the batch mean. Deterministic (no atomics).
// ---------------------------------------------------------------------------
__global__ __launch_bounds__(TPB) void mixit_reduce_kernel(
    const float* __restrict__ partials,   // (BATCH*SPLIT, 256)
    float* __restrict__ out)              // scalar
{
    __shared__ float gram[256];
    __shared__ float red[256];

    const int tid = threadIdx.x;
    float batch_sum = 0.f;   // meaningful on thread 0 only

    for (int b = 0; b < BATCH; ++b) {
        float g = 0.f;
        #pragma unroll
        for (int s = 0; s < SPLIT; ++s)
            g += partials[(size_t)(b * SPLIT + s) * 256 + tid];
        gram[tid] = g;
        __syncthreads();

        float val = INFINITY;
        if (tid >= 1 && tid <= 254)
            val = combo_loss(gram, tid);

        red[tid] = val;
        __syncthreads();
        #pragma unroll
        for (int s = TPB / 2; s > 0; s >>= 1) {
            if (tid < s) red[tid] = fminf(red[tid], red[tid + s]);
            __syncthreads();
        }
        if (tid == 0) batch_sum += red[0];
        __syncthreads();   // protect gram/red before next iteration
    }

    if (tid == 0) out[0] = batch_sum * (1.0f / BATCH);
}

extern "C" void kernel_launch(void* const* d_in, const int* in_sizes, int n_in,
                              void* d_out, int out_size, void* d_ws, size_t ws_size,
                              hipStream_t stream)
{
    const float* est = (const float*)d_in[0];   // (B, M, T) fp32
    const float* mix = (const float*)d_in[1];   // (B, 2, T) fp32
    float* out = (float*)d_out;                 // scalar fp32
    float* ws  = (float*)d_ws;                  // >= BATCH*SPLIT*256 floats (64 KB)

    const int T = in_sizes[1] / (2 * BATCH);    // 32000

    mixit_gram_kernel<<<BATCH * SPLIT, TPB, 0, stream>>>(est, mix, ws, T);
    mixit_reduce_kernel<<<1, TPB, 0, stream>>>(ws, out);
}